// HyperGraphAttentionLayerSparse_63118839382177
// MI455X (gfx1250) — compile-verified
//
#include <hip/hip_runtime.h>
#include <hip/hip_bf16.h>

#define NN   20000      // nodes
#define EE   5000       // hyperedges
#define NZ   160000     // incidence pairs
#define D    128        // feature dim (D_IN == D_OUT)
#define EPSF 1e-10f
#define ALPHAF 0.2f

typedef float v2f __attribute__((ext_vector_type(2)));
typedef float v8f __attribute__((ext_vector_type(8)));

// ---------------------------------------------------------------- utilities
__global__ void zero_f_k(float* p, long n) {
    long i = (long)blockIdx.x * blockDim.x + threadIdx.x;
    long st = (long)gridDim.x * blockDim.x;
    for (; i < n; i += st) p[i] = 0.0f;
}

// ---------------------------------------------------------------- degrees
__global__ void degrees_k(const int* __restrict__ rows, const int* __restrict__ cols,
                          const float* __restrict__ vals,
                          float* __restrict__ dv, float* __restrict__ de,
                          int* __restrict__ cnt) {
    int i = blockIdx.x * blockDim.x + threadIdx.x;
    if (i >= NZ) return;
    atomicAdd(&dv[rows[i]], vals[i]);
    atomicAdd(&de[cols[i]], vals[i]);
    atomicAdd(&cnt[rows[i]], 1);
}

__global__ void invdeg_k(const float* __restrict__ dv, const float* __restrict__ de,
                         float* __restrict__ dvi, float* __restrict__ dei) {
    int i = blockIdx.x * blockDim.x + threadIdx.x;
    if (i < NN) dvi[i] = rsqrtf(dv[i] + EPSF);
    if (i < EE) dei[i] = 1.0f / (de[i] + EPSF);
}

// ---------------------------------------------------------------- X_proj = x @ weight  (fp32 WMMA)
// grid.x = NN/16 tiles of rows; 8 waves per block, wave w owns output cols [16w,16w+16)
__global__ __launch_bounds__(256) void gemm_xw_k(const float* __restrict__ x,
                                                 const float* __restrict__ w,
                                                 float* __restrict__ xp) {
    __shared__ float sW[D * D];   // 64 KB: full weight tile (K-major, row k = D cols)
    __shared__ float sX[16 * D];  // 8 KB:  16-row x tile

    const int t = threadIdx.x;
    const int m_base = blockIdx.x * 16;

    for (int i = t * 4; i < D * D; i += 256 * 4)
        *(float4*)&sW[i] = *(const float4*)&w[i];
    for (int i = t * 4; i < 16 * D; i += 256 * 4) {
        int r = i >> 7, c = i & (D - 1);
        *(float4*)&sX[i] = *(const float4*)&x[(size_t)(m_base + r) * D + c];
    }
    __syncthreads();

    const int wave = t >> 5;
    const int lane = t & 31;
    const int half = lane >> 4;        // lanes 16-31 handle K offset +2 (and M rows 8-15 of C)
    const int l    = lane & 15;
    const int n_base = wave * 16;

    v8f c = {};
    for (int k0 = 0; k0 < D; k0 += 4) {
        const int k = k0 + half * 2;
        v2f a, b;
        a.x = sX[l * D + k];
        a.y = sX[l * D + k + 1];
        b.x = sW[k * D + n_base + l];
        b.y = sW[(k + 1) * D + n_base + l];
        c = __builtin_amdgcn_wmma_f32_16x16x4_f32(false, a, false, b,
                                                  (short)0, c, false, false);
    }

    const int row0 = m_base + half * 8;
    const int col  = n_base + l;
#pragma unroll
    for (int r = 0; r < 8; ++r)
        xp[(size_t)(row0 + r) * D + col] = c[r];
}

// ---------------------------------------------------------------- E_feat scatter (node -> edge)
// 32 lanes per pair, 4 dims per lane
__global__ void scatter_edge_k(const int* __restrict__ rows, const int* __restrict__ cols,
                               const float* __restrict__ vals,
                               const float* __restrict__ xp, const float* __restrict__ dvi,
                               float* __restrict__ ef) {
    long tid  = (long)blockIdx.x * blockDim.x + threadIdx.x;
    long pair = tid >> 5;
    int  d4   = (int)(tid & 31) * 4;
    if (pair >= NZ) return;
    int r = rows[pair], cE = cols[pair];
    float sc = vals[pair] * dvi[r];
    float4 v = *(const float4*)&xp[(size_t)r * D + d4];
    float* dst = &ef[(size_t)cE * D + d4];
    atomicAdd(dst + 0, sc * v.x); atomicAdd(dst + 1, sc * v.y);
    atomicAdd(dst + 2, sc * v.z); atomicAdd(dst + 3, sc * v.w);
}

// ---------------------------------------------------------------- F_out scatter (edge -> node)
__global__ void scatter_node_k(const int* __restrict__ rows, const int* __restrict__ cols,
                               const float* __restrict__ vals,
                               const float* __restrict__ ef, const float* __restrict__ dei,
                               float* __restrict__ fo) {
    long tid  = (long)blockIdx.x * blockDim.x + threadIdx.x;
    long pair = tid >> 5;
    int  d4   = (int)(tid & 31) * 4;
    if (pair >= NZ) return;
    int r = rows[pair], cE = cols[pair];
    float sc = vals[pair] * dei[cE];
    float4 v = *(const float4*)&ef[(size_t)cE * D + d4];
    float* dst = &fo[(size_t)r * D + d4];
    atomicAdd(dst + 0, sc * v.x); atomicAdd(dst + 1, sc * v.y);
    atomicAdd(dst + 2, sc * v.z); atomicAdd(dst + 3, sc * v.w);
}

// ---------------------------------------------------------------- Y_hat = F_out * dv_inv + X_proj
__global__ void yhat_k(const float* __restrict__ fo, const float* __restrict__ xp,
                       const float* __restrict__ dvi, float* __restrict__ yh) {
    long i = (long)blockIdx.x * blockDim.x + threadIdx.x;
    if (i >= (long)NN * D) return;
    int n = (int)(i >> 7);
    yh[i] = fo[i] * dvi[n] + xp[i];
}

// ---------------------------------------------------------------- per-pair logits + Z accumulation
// wave per pair: s = leakyrelu(Yhat[row].a[0:128] + Yhat[col].a[128:256])
__global__ __launch_bounds__(256) void logits_k(const int* __restrict__ rows,
                                                const int* __restrict__ cols,
                                                const float* __restrict__ yh,
                                                const float* __restrict__ a,
                                                float* __restrict__ s,
                                                float* __restrict__ zsum) {
    int pair = (int)(((long)blockIdx.x * blockDim.x + threadIdx.x) >> 5);
    int lane = threadIdx.x & 31;
    if (pair >= NZ) return;
    int r = rows[pair], cE = cols[pair];
    int d = lane * 4;
    float4 hv = *(const float4*)&yh[(size_t)r  * D + d];
    float4 hu = *(const float4*)&yh[(size_t)cE * D + d];   // faithful bug: cols index node tensor
    float4 a0 = *(const float4*)&a[d];
    float4 a1 = *(const float4*)&a[D + d];
    float p = hv.x * a0.x + hv.y * a0.y + hv.z * a0.z + hv.w * a0.w
            + hu.x * a1.x + hu.y * a1.y + hu.z * a1.z + hu.w * a1.w;
    for (int off = 16; off; off >>= 1) p += __shfl_xor(p, off, 32);
    if (lane == 0) {
        float e = p > 0.0f ? p : ALPHAF * p;
        s[pair] = e;
        atomicAdd(&zsum[r], expf(e));
    }
}

// Z_n = (E - cnt_n)*exp(0) + sum exp(s);  store 1/Z
__global__ void zinv_k(const float* __restrict__ zsum, const int* __restrict__ cnt,
                       float* __restrict__ zi) {
    int n = blockIdx.x * blockDim.x + threadIdx.x;
    if (n >= NN) return;
    zi[n] = 1.0f / ((float)EE - (float)cnt[n] + zsum[n]);
}

// ---------------------------------------------------------------- g[d] = sum_n zi[n]*X_proj[n,d]
__global__ __launch_bounds__(128) void gvec_k(const float* __restrict__ xp,
                                              const float* __restrict__ zi,
                                              float* __restrict__ g) {
    int d  = threadIdx.x;
    int n0 = blockIdx.x * 64;
    int n1 = n0 + 64; if (n1 > NN) n1 = NN;
    float acc = 0.0f;
    for (int n = n0; n < n1; ++n) acc += zi[n] * xp[(size_t)n * D + d];
    atomicAdd(&g[d], acc);
}

// edge_feats[e,:] = g (the uniform softmax baseline applied to every column)
__global__ void edge_init_k(const float* __restrict__ g, float* __restrict__ efat) {
    long i = (long)blockIdx.x * blockDim.x + threadIdx.x;
    if (i >= (long)EE * D) return;
    efat[i] = g[i & (D - 1)];
}

// edge_feats[col] += ((exp(s)-1)/Z_row) * X_proj[row]
__global__ void scatter_delta_k(const int* __restrict__ rows, const int* __restrict__ cols,
                                const float* __restrict__ s, const float* __restrict__ zi,
                                const float* __restrict__ xp, float* __restrict__ efat) {
    long tid  = (long)blockIdx.x * blockDim.x + threadIdx.x;
    long pair = tid >> 5;
    int  d4   = (int)(tid & 31) * 4;
    if (pair >= NZ) return;
    int r = rows[pair], cE = cols[pair];
    float sc = (expf(s[pair]) - 1.0f) * zi[r];
    float4 v = *(const float4*)&xp[(size_t)r * D + d4];
    float* dst = &efat[(size_t)cE * D + d4];
    atomicAdd(dst + 0, sc * v.x); atomicAdd(dst + 1, sc * v.y);
    atomicAdd(dst + 2, sc * v.z); atomicAdd(dst + 3, sc * v.w);
}

// out init with bias, then out[row] += vals * edge_feats[col]
__global__ void bias_k(const float* __restrict__ bias, float* __restrict__ out) {
    long i = (long)blockIdx.x * blockDim.x + threadIdx.x;
    if (i >= (long)NN * D) return;
    out[i] = bias[i & (D - 1)];
}

__global__ void scatter_out_k(const int* __restrict__ rows, const int* __restrict__ cols,
                              const float* __restrict__ vals,
                              const float* __restrict__ efat, float* __restrict__ out) {
    long tid  = (long)blockIdx.x * blockDim.x + threadIdx.x;
    long pair = tid >> 5;
    int  d4   = (int)(tid & 31) * 4;
    if (pair >= NZ) return;
    int r = rows[pair], cE = cols[pair];
    float sc = vals[pair];
    float4 v = *(const float4*)&efat[(size_t)cE * D + d4];
    float* dst = &out[(size_t)r * D + d4];
    atomicAdd(dst + 0, sc * v.x); atomicAdd(dst + 1, sc * v.y);
    atomicAdd(dst + 2, sc * v.z); atomicAdd(dst + 3, sc * v.w);
}

// ---------------------------------------------------------------- host side
extern "C" void kernel_launch(void* const* d_in, const int* in_sizes, int n_in,
                              void* d_out, int out_size, void* d_ws, size_t ws_size,
                              hipStream_t stream) {
    const float* x    = (const float*)d_in[0];
    const int*   rows = (const int*)  d_in[1];
    const int*   cols = (const int*)  d_in[2];
    const float* vals = (const float*)d_in[3];
    const float* w    = (const float*)d_in[4];
    const float* a    = (const float*)d_in[5];
    const float* bias = (const float*)d_in[6];
    float* out = (float*)d_out;

    // workspace layout (floats)
    float* ws   = (float*)d_ws;
    float* dv   = ws;                    // NN
    float* de   = dv   + NN;             // EE
    float* dvi  = de   + EE;             // NN
    float* dei  = dvi  + NN;             // EE
    float* xp   = dei  + EE;             // NN*D
    float* ef   = xp   + (size_t)NN * D; // EE*D   (E_feat)
    float* fo   = ef   + (size_t)EE * D; // NN*D   (F_out)
    float* yh   = fo   + (size_t)NN * D; // NN*D   (Y_hat)
    float* sarr = yh   + (size_t)NN * D; // NZ
    float* zsum = sarr + NZ;             // NN
    float* zi   = zsum + NN;             // NN
    float* g    = zi   + NN;             // D
    float* efat = g    + D;              // EE*D   (edge_feats)
    int*   cnt  = (int*)(efat + (size_t)EE * D);  // NN (ints; just past the zeroed range)

    const long total_ws_f = (long)(3 * NN + 2 * EE) + (long)3 * NN * D
                          + (long)2 * EE * D + NZ + D + NN;

    const int T = 256;
    zero_f_k<<<2048, T, 0, stream>>>(ws, total_ws_f);

    degrees_k<<<(NZ + T - 1) / T, T, 0, stream>>>(rows, cols, vals, dv, de, cnt);
    invdeg_k<<<(NN + T - 1) / T, T, 0, stream>>>(dv, de, dvi, dei);

    gemm_xw_k<<<NN / 16, T, 0, stream>>>(x, w, xp);

    const int pair_blocks = (int)(((long)NZ * 32 + T - 1) / T);   // 20000
    scatter_edge_k<<<pair_blocks, T, 0, stream>>>(rows, cols, vals, xp, dvi, ef);
    scatter_node_k<<<pair_blocks, T, 0, stream>>>(rows, cols, vals, ef, dei, fo);

    yhat_k<<<(int)(((long)NN * D + T - 1) / T), T, 0, stream>>>(fo, xp, dvi, yh);

    logits_k<<<pair_blocks, T, 0, stream>>>(rows, cols, yh, a, sarr, zsum);
    zinv_k<<<(NN + T - 1) / T, T, 0, stream>>>(zsum, cnt, zi);

    gvec_k<<<(NN + 63) / 64, 128, 0, stream>>>(xp, zi, g);
    edge_init_k<<<(int)(((long)EE * D + T - 1) / T), T, 0, stream>>>(g, efat);
    scatter_delta_k<<<pair_blocks, T, 0, stream>>>(rows, cols, sarr, zi, xp, efat);

    bias_k<<<(int)(((long)NN * D + T - 1) / T), T, 0, stream>>>(bias, out);
    scatter_out_k<<<pair_blocks, T, 0, stream>>>(rows, cols, vals, efat, out);
}